// CircuitTorchWrapper_44865228374112
// MI455X (gfx1250) — compile-verified
//
#include <hip/hip_runtime.h>

#define N_QUBIT 22
#define DIM (1u << N_QUBIT)
#define N_LAYER 8
#define NSEG 8            // 3 "HI" segments (qubits 0..8) + 5 "LO" segments (qubits 8..21)
#define MAT_ELEMS 256     // 16x16 f32 per segment matrix

typedef float v2f __attribute__((ext_vector_type(2)));
typedef float v8f __attribute__((ext_vector_type(8)));

// ---------------------------------------------------------------------------
// Segment tables. Segment s acts on 4 adjacent qubits q[s][0..3] (q[s][0] is
// the MSB of the 4-bit basis index). ry mask bit i => apply Ry(theta[q_i]) at
// group position i; cn mask bit j => CNOT(pos j -> pos j+1), applied j = 0,1,2
// ascending (matching the reference ladder order).
//   HI:  s0: Ry(0..3)+C(0,1)(1,2)(2,3)   s1: Ry(4..6)+C(3,4)(4,5)(5,6)
//        s2: Ry(7,8)+C(6,7)(7,8)  (embedded on qubits 5..8, identity on 5,6)
//   LO:  s3..s6: Ry(a+1..a+3)+C(a..a+2) for a=8,11,14,17
//        s7: Ry(21)+C(20,21)      (embedded on qubits 18..21)
// ---------------------------------------------------------------------------
__device__ const int g_seg_q[NSEG][4] = {
    {0, 1, 2, 3},     {3, 4, 5, 6},     {5, 6, 7, 8},
    {8, 9, 10, 11},   {11, 12, 13, 14}, {14, 15, 16, 17},
    {17, 18, 19, 20}, {18, 19, 20, 21}};
__device__ const int g_seg_ry[NSEG] = {0xF, 0xE, 0xC, 0xE, 0xE, 0xE, 0xE, 0x8};
__device__ const int g_seg_cn[NSEG] = {0x7, 0x7, 0x6, 0x7, 0x7, 0x7, 0x7, 0x4};

// Build, for each layer, the 8 fused 16x16 segment matrices from theta.
// mats layout: [layer][seg][row][col] f32.
__global__ __launch_bounds__(256) void prep_mats(const float* __restrict__ theta,
                                                 float* __restrict__ mats) {
  const int l = blockIdx.x;            // layer
  const int tid = threadIdx.x;         // one (r,c) entry per thread
  const int r = tid >> 4;
  const int c = tid & 15;
  for (int s = 0; s < NSEG; ++s) {
    // R[r][c] = prod_i f_i(r_i, c_i), f_i = Ry or I. pos i <-> bit (3-i).
    float v = 1.0f;
#pragma unroll
    for (int i = 0; i < 4; ++i) {
      const int ri = (r >> (3 - i)) & 1;
      const int ci = (c >> (3 - i)) & 1;
      if ((g_seg_ry[s] >> i) & 1) {
        const float th = 0.5f * theta[l * N_QUBIT + g_seg_q[s][i]];
        const float cs = cosf(th);
        const float sn = sinf(th);
        // Ry = [[c,-s],[s,c]]
        v *= ri ? (ci ? cs : sn) : (ci ? -sn : cs);
      } else if (ri != ci) {
        v = 0.0f;
      }
    }
    // CNOT sub-chain = row permutation: posbit(j+1) ^= posbit(j), j ascending.
    int pr = r;
    const int cn = g_seg_cn[s];
    if (cn & 1) pr ^= ((pr >> 3) & 1) << 2;
    if (cn & 2) pr ^= ((pr >> 2) & 1) << 1;
    if (cn & 4) pr ^= ((pr >> 1) & 1);
    mats[((l * NSEG + s) * 16 + pr) * 16 + c] = v;
  }
}

// ---------------------------------------------------------------------------
// Apply one 16x16 segment matrix U to the 16k-float LDS tile, where the 4-bit
// basis field sits at LDS-index bit position p. Each wave owns 8 disjoint
// 16-column tiles; per tile: gather B (4x16 x4 K-chunks), 4 chained
// v_wmma_f32_16x16x4_f32, scatter D back in place.
// The tile loop trip count is a literal and `wave` is an SGPR, so the loop is
// a scalar branch and EXEC stays all-ones across every WMMA (ISA §7.12).
// ISA operand layouts (32-bit):
//   A 16x4 : lanes 0-15 M=0..15 K{0,1}; lanes 16-31 M=0..15 K{2,3}
//   B 4x16 : lanes 0-15 N=0..15 K{0,1}; lanes 16-31 N=0..15 K{2,3}
//   C/D    : VGPR j: lanes 0-15 M=j, lanes 16-31 M=j+8; N = lane&15
// ---------------------------------------------------------------------------
__device__ __forceinline__ void apply_seg(float* lds, const float* __restrict__ U,
                                          const int p, const int lane,
                                          const int wave) {
  const int arow = lane & 15;
  const int khalf = lane >> 4;  // 0: K{k0,k0+1}, 1: K{k0+2,k0+3}
  v2f a[4];
#pragma unroll
  for (int kc = 0; kc < 4; ++kc) {
    const int k0 = kc * 4 + khalf * 2;
    a[kc].x = U[arow * 16 + k0];
    a[kc].y = U[arow * 16 + k0 + 1];
  }
  const int lomask = (1 << p) - 1;
#pragma unroll 1
  for (int tt = 0; tt < 8; ++tt) {        // 64 column-tiles, 8 waves, 8 each
    const int t = (tt << 3) + wave;       // uniform trip count, sgpr wave id
    const int c = t * 16 + arow;          // this lane's column
    const int base = ((c >> p) << (p + 4)) | (c & lomask);
    v2f b[4];
#pragma unroll
    for (int kc = 0; kc < 4; ++kc) {
      const int k0 = kc * 4 + khalf * 2;
      b[kc].x = lds[base + (k0 << p)];
      b[kc].y = lds[base + ((k0 + 1) << p)];
    }
    v8f acc = {0.f, 0.f, 0.f, 0.f, 0.f, 0.f, 0.f, 0.f};
#pragma unroll
    for (int kc = 0; kc < 4; ++kc) {
      acc = __builtin_amdgcn_wmma_f32_16x16x4_f32(
          /*neg_a=*/false, a[kc], /*neg_b=*/false, b[kc],
          /*c_mod=*/(short)0, acc, /*reuse_a=*/false, /*reuse_b=*/false);
    }
#pragma unroll
    for (int j = 0; j < 8; ++j) {
      lds[base + ((j + khalf * 8) << p)] = acc[j];
    }
  }
}

// HI kernel: qubits 0..8 (flat bits 21..13). Block = 512 h-values x 32
// consecutive low addresses => 64 KB LDS tile; segments at LDS bits 10,7,5.
__global__ __launch_bounds__(256) void hi_kernel(const float* __restrict__ in,
                                                 float* __restrict__ out,
                                                 const float* __restrict__ mats,
                                                 const int layer) {
  __shared__ float lds[16384];
  const float* M = mats + layer * NSEG * MAT_ELEMS;  // segs 0..2
  const int tid = threadIdx.x;
  const int lbase = blockIdx.x * 32;
  // 128-bit staging: i is 4-aligned inside a 32-chunk, so the 4 floats are
  // contiguous both in LDS and in global memory.
#pragma unroll 1
  for (int i = tid * 4; i < 16384; i += 1024) {
    *(float4*)(lds + i) =
        *(const float4*)(in + (i >> 5) * 8192 + lbase + (i & 31));
  }
  __syncthreads();
  const int lane = tid & 31;
  const int wave = __builtin_amdgcn_readfirstlane(tid >> 5);
  const int pv[3] = {10, 7, 5};
#pragma unroll 1
  for (int s = 0; s < 3; ++s) {
    apply_seg(lds, M + s * MAT_ELEMS, pv[s], lane, wave);
    __syncthreads();
  }
#pragma unroll 1
  for (int i = tid * 4; i < 16384; i += 1024) {
    *(float4*)(out + (i >> 5) * 8192 + lbase + (i & 31)) =
        *(const float4*)(lds + i);
  }
}

// LO kernel: qubits 8..21 (flat bits 13..0). Block = 16384 consecutive floats;
// segments at LDS bits 10,7,4,1,0.
__global__ __launch_bounds__(256) void lo_kernel(const float* __restrict__ in,
                                                 float* __restrict__ out,
                                                 const float* __restrict__ mats,
                                                 const int layer) {
  __shared__ float lds[16384];
  const float* M = mats + (layer * NSEG + 3) * MAT_ELEMS;  // segs 3..7
  const int tid = threadIdx.x;
  const unsigned base = blockIdx.x * 16384u;
#pragma unroll 1
  for (int i = tid * 4; i < 16384; i += 1024) {
    *(float4*)(lds + i) = *(const float4*)(in + base + i);
  }
  __syncthreads();
  const int lane = tid & 31;
  const int wave = __builtin_amdgcn_readfirstlane(tid >> 5);
  const int pv[5] = {10, 7, 4, 1, 0};
#pragma unroll 1
  for (int s = 0; s < 5; ++s) {
    apply_seg(lds, M + s * MAT_ELEMS, pv[s], lane, wave);
    __syncthreads();
  }
#pragma unroll 1
  for (int i = tid * 4; i < 16384; i += 1024) {
    *(float4*)(out + base + i) = *(const float4*)(lds + i);
  }
}

__global__ __launch_bounds__(256) void zero_kernel(float* __restrict__ p) {
  const unsigned i = (blockIdx.x * 256u + threadIdx.x) * 4u;
  *(float4*)(p + i) = make_float4(0.f, 0.f, 0.f, 0.f);
}

extern "C" void kernel_launch(void* const* d_in, const int* in_sizes, int n_in,
                              void* d_out, int out_size, void* d_ws, size_t ws_size,
                              hipStream_t stream) {
  const float* theta = (const float*)d_in[0];  // [8][22] f32
  const float* q0 = (const float*)d_in[1];     // [DIM] f32
  float* out = (float*)d_out;                  // [2][DIM] f32
  float* mats = (float*)d_ws;                  // 8*8*256 f32 = 64 KB
  float* psi = out;                            // real half doubles as state

  prep_mats<<<N_LAYER, 256, 0, stream>>>(theta, mats);
  zero_kernel<<<DIM / 1024, 256, 0, stream>>>(out + DIM);  // imag part == 0

  for (int l = 0; l < N_LAYER; ++l) {
    hi_kernel<<<256, 256, 0, stream>>>(l == 0 ? q0 : psi, psi, mats, l);
    lo_kernel<<<256, 256, 0, stream>>>(psi, psi, mats, l);
  }
}